// StateSpaceModel_46188078301336
// MI455X (gfx1250) — compile-verified
//
#include <hip/hip_runtime.h>

typedef __attribute__((ext_vector_type(2))) float v2f;
typedef __attribute__((ext_vector_type(8))) float v8f;

#define BATCH 8
#define LEN   8192
#define DIN   128
#define NST   128
#define DOUT  128
#define BL    (BATCH * LEN)   /* 65536 rows */
#define LC    128             /* chunk length   */
#define NC    (LEN / LC)      /* 64 chunks      */

__device__ __forceinline__ v8f wmma_f32(v2f a, v2f b, v8f c) {
  // V_WMMA_F32_16X16X4_F32 : D = A(16x4) * B(4x16) + C(16x16), fp32
  return __builtin_amdgcn_wmma_f32_16x16x4_f32(
      /*neg_a=*/false, a, /*neg_b=*/false, b,
      /*c_mod=*/(short)0, c, /*reuse_a=*/false, /*reuse_b=*/false);
}

// ---------------------------------------------------------------------------
// K0: prep. Transpose B_re/B_im/C_re/C_im (so GEMM B-fragments become one
// contiguous b64 load) and build power table P[tloc][n] = A_n^(tloc+1),
// tloc = 0..127. All outputs are tiny and stay L2-resident.
// ---------------------------------------------------------------------------
__global__ void k0_prep(const float* __restrict__ Are,
                        const float* __restrict__ Aim,
                        const float* __restrict__ Bre,
                        const float* __restrict__ Bim,
                        const float* __restrict__ Cre,
                        const float* __restrict__ Cim,
                        float* __restrict__ Btre, float* __restrict__ Btim,
                        float* __restrict__ Ctre, float* __restrict__ Ctim,
                        float* __restrict__ Pre,  float* __restrict__ Pim) {
  const int idx = blockIdx.x * blockDim.x + threadIdx.x;  // 128*128 = 16384
  const int i = idx >> 7;        // source row (k for B/C, tloc for P)
  const int j = idx & 127;       // source col (n)
  Btre[j * DIN + i] = Bre[i * NST + j];
  Btim[j * DIN + i] = Bim[i * NST + j];
  Ctre[j * NST + i] = Cre[i * DOUT + j];
  Ctim[j * NST + i] = Cim[i * DOUT + j];
  // P[i][j] = A_j^(i+1) via binary exponentiation (<= 8 steps)
  float br = Are[j], bi = Aim[j];
  float wr = 1.f, wi = 0.f;
  int e = i + 1;
  while (e) {
    if (e & 1) { const float nr = wr * br - wi * bi; wi = wr * bi + wi * br; wr = nr; }
    e >>= 1;
    if (e)     { const float nr = br * br - bi * bi; bi = 2.f * br * bi; br = nr; }
  }
  Pre[i * NST + j] = wr;
  Pim[i * NST + j] = wi;
}

// ---------------------------------------------------------------------------
// K1: uB = u @ (B_re + i B_im)   -- fp32 WMMA, M=BL, N=NST, K=DIN
// one wave per 16x16 output tile; 8 waves/block cover N=128. B transposed.
// ---------------------------------------------------------------------------
__global__ void k1_in_proj(const float* __restrict__ u,
                           const float* __restrict__ Btre,
                           const float* __restrict__ Btim,
                           float* __restrict__ uBre,
                           float* __restrict__ uBim) {
  const int lane = threadIdx.x & 31;
  const int wave = threadIdx.x >> 5;
  const int half = lane >> 4;      // selects K pair
  const int l16  = lane & 15;      // M row (A) / N col (B,D)
  const size_t m0 = (size_t)blockIdx.x * 16;
  const int n0 = wave * 16;

  v8f accRe = {};
  v8f accIm = {};
  const float* uRow = u + (m0 + l16) * DIN;
  const float* btr = Btre + (size_t)(n0 + l16) * DIN;   // Bt[n][k]
  const float* bti = Btim + (size_t)(n0 + l16) * DIN;

  for (int k0 = 0; k0 < DIN; k0 += 4) {
    const int ka = k0 + 2 * half;
    v2f a  = *(const v2f*)(uRow + ka);   // contiguous K pair
    v2f br = *(const v2f*)(btr + ka);    // (B[ka][n], B[ka+1][n])
    v2f bi = *(const v2f*)(bti + ka);
    accRe = wmma_f32(a, br, accRe);
    accIm = wmma_f32(a, bi, accIm);
  }

  for (int r = 0; r < 8; ++r) {
    const size_t row = m0 + r + 8 * half;   // C/D layout: VGPR r -> M=r / M=r+8
    uBre[row * NST + n0 + l16] = accRe[r];
    uBim[row * NST + n0 + l16] = accIm[r];
  }
}

// ---------------------------------------------------------------------------
// K2: chunk-local scan  x[t] = A*x[t-1] + uB[t]  within each LC=128 chunk.
// thread per (batch, chunk, n); coalesced over n; in-place, emits chunk finals
// ---------------------------------------------------------------------------
__global__ void k2_chunk_scan(const float* __restrict__ Are,
                              const float* __restrict__ Aim,
                              float* __restrict__ xre,
                              float* __restrict__ xim,
                              float* __restrict__ cfre,
                              float* __restrict__ cfim) {
  const int idx = blockIdx.x * blockDim.x + threadIdx.x;  // BATCH*NC*NST = 65536
  const int n = idx & (NST - 1);
  const int c = (idx >> 7) & (NC - 1);
  const int b = idx >> 13;
  const float ar = Are[n], ai = Aim[n];
  float sr = 0.f, si = 0.f;
  const size_t base = ((size_t)b * LEN + (size_t)c * LC) * NST + n;
  for (int t = 0; t < LC; ++t) {
    const size_t off = base + (size_t)t * NST;
    const float ur = xre[off], ui = xim[off];
    const float nr = fmaf(ar, sr, fmaf(-ai, si, ur));
    const float ni = fmaf(ar, si, fmaf(ai, sr, ui));
    sr = nr; si = ni;
    xre[off] = sr;
    xim[off] = si;
  }
  const int o = (b * NC + c) * NST + n;
  cfre[o] = sr;
  cfim[o] = si;
}

// ---------------------------------------------------------------------------
// K3: inter-chunk carry scan: S_{c+1} = A^LC * S_c + final_c ; S_0 = 0.
// Also writes x[:, L-1] (== S_NC) to the tail of d_out, interleaved complex.
// ---------------------------------------------------------------------------
__global__ void k3_carry(const float* __restrict__ Are,
                         const float* __restrict__ Aim,
                         const float* __restrict__ cfre,
                         const float* __restrict__ cfim,
                         float* __restrict__ care,
                         float* __restrict__ caim,
                         float* __restrict__ out_xlast) {
  const int idx = blockIdx.x * blockDim.x + threadIdx.x;  // BATCH*NST = 1024
  const int n = idx & (NST - 1);
  const int b = idx >> 7;
  float pr = Are[n], pi = Aim[n];
  for (int i = 0; i < 7; ++i) {                       // A^128 by squaring
    const float nr = pr * pr - pi * pi;
    const float ni = 2.f * pr * pi;
    pr = nr; pi = ni;
  }
  float Sr = 0.f, Si = 0.f;
  for (int c = 0; c < NC; ++c) {
    const int o = (b * NC + c) * NST + n;
    care[o] = Sr;
    caim[o] = Si;
    const float fr = cfre[o], fi = cfim[o];
    const float nr = fmaf(pr, Sr, fmaf(-pi, Si, fr));
    const float ni = fmaf(pr, Si, fmaf(pi, Sr, fi));
    Sr = nr; Si = ni;
  }
  out_xlast[(b * NST + n) * 2 + 0] = Sr;              // complex64 -> interleaved
  out_xlast[(b * NST + n) * 2 + 1] = Si;
}

// ---------------------------------------------------------------------------
// K5: fused carry-apply + output projection.
// Each lane's A-fragment row is a fixed time t, K index is state n, so the
// carry term A_n^{tloc+1} * S_c(b,n) is applied on the fly from the L2-resident
// P table and carry array:   x_full = x_local + P(tloc,:) (.) S_c(b,:)
// then  y = x_re @ C_re - x_im @ C_im  via fp32 WMMA.
// ---------------------------------------------------------------------------
__global__ void k5_out_fused(const float* __restrict__ xre,
                             const float* __restrict__ xim,
                             const float* __restrict__ Ctre,
                             const float* __restrict__ Ctim,
                             const float* __restrict__ Pre,
                             const float* __restrict__ Pim,
                             const float* __restrict__ care,
                             const float* __restrict__ caim,
                             float* __restrict__ y) {
  const int lane = threadIdx.x & 31;
  const int wave = threadIdx.x >> 5;
  const int half = lane >> 4;
  const int l16  = lane & 15;
  const size_t m0 = (size_t)blockIdx.x * 16;
  const int n0 = wave * 16;

  const size_t row = m0 + l16;             // global time-row of this lane
  const int t    = (int)(row & (LEN - 1));
  const int b    = (int)(row >> 13);
  const int tloc = t & (LC - 1);
  const int c    = t >> 7;

  const float* xr  = xre + row * NST;
  const float* xi  = xim + row * NST;
  const float* Pr  = Pre + (size_t)tloc * NST;
  const float* Pi  = Pim + (size_t)tloc * NST;
  const float* Sr  = care + (size_t)(b * NC + c) * NST;
  const float* Si  = caim + (size_t)(b * NC + c) * NST;
  const float* ctr = Ctre + (size_t)(n0 + l16) * NST;  // Ct[n][k]
  const float* cti = Ctim + (size_t)(n0 + l16) * NST;

  v8f accR = {};
  v8f accI = {};

  for (int k0 = 0; k0 < NST; k0 += 4) {
    const int ka = k0 + 2 * half;
    v2f ar = *(const v2f*)(xr + ka);
    v2f ai = *(const v2f*)(xi + ka);
    v2f pr = *(const v2f*)(Pr + ka);
    v2f pi = *(const v2f*)(Pi + ka);
    v2f sr = *(const v2f*)(Sr + ka);
    v2f si = *(const v2f*)(Si + ka);
    // x_full = x_local + P * S   (complex, componentwise over the K pair)
    v2f xrf = ar + pr * sr - pi * si;
    v2f xif = ai + pr * si + pi * sr;
    v2f cr = *(const v2f*)(ctr + ka);    // (C[ka][n], C[ka+1][n])
    v2f ci = *(const v2f*)(cti + ka);
    accR = wmma_f32(xrf, cr, accR);
    accI = wmma_f32(xif, ci, accI);
  }

  const v8f res = accR - accI;
  for (int r = 0; r < 8; ++r) {
    const size_t orow = m0 + r + 8 * half;
    y[orow * DOUT + n0 + l16] = res[r];
  }
}

// ---------------------------------------------------------------------------
extern "C" void kernel_launch(void* const* d_in, const int* in_sizes, int n_in,
                              void* d_out, int out_size, void* d_ws, size_t ws_size,
                              hipStream_t stream) {
  const float* u   = (const float*)d_in[0];
  const float* Are = (const float*)d_in[1];
  const float* Aim = (const float*)d_in[2];
  const float* Bre = (const float*)d_in[3];
  const float* Bim = (const float*)d_in[4];
  const float* Cre = (const float*)d_in[5];
  const float* Cim = (const float*)d_in[6];

  float* y     = (float*)d_out;
  float* xlast = y + (size_t)BL * DOUT;               // [8,128] complex64 tail

  float* ws   = (float*)d_ws;
  float* xre  = ws;                                   // BL*NST (uB_re -> x_re)
  float* xim  = xre + (size_t)BL * NST;               // BL*NST (uB_im -> x_im)
  float* cfre = xim + (size_t)BL * NST;               // chunk finals
  float* cfim = cfre + (size_t)BATCH * NC * NST;
  float* care = cfim + (size_t)BATCH * NC * NST;      // carries (state before chunk)
  float* caim = care + (size_t)BATCH * NC * NST;
  float* Btre = caim + (size_t)BATCH * NC * NST;      // transposed weights
  float* Btim = Btre + (size_t)NST * DIN;
  float* Ctre = Btim + (size_t)NST * DIN;
  float* Ctim = Ctre + (size_t)NST * DOUT;
  float* Pre  = Ctim + (size_t)NST * DOUT;            // P[tloc][n] = A_n^(tloc+1)
  float* Pim  = Pre + (size_t)LC * NST;

  k0_prep<<<(128 * 128) / 256, 256, 0, stream>>>(Are, Aim, Bre, Bim, Cre, Cim,
                                                 Btre, Btim, Ctre, Ctim, Pre, Pim);
  k1_in_proj<<<BL / 16, 256, 0, stream>>>(u, Btre, Btim, xre, xim);
  k2_chunk_scan<<<(BATCH * NC * NST) / 256, 256, 0, stream>>>(Are, Aim, xre, xim, cfre, cfim);
  k3_carry<<<(BATCH * NST) / 256, 256, 0, stream>>>(Are, Aim, cfre, cfim, care, caim, xlast);
  k5_out_fused<<<BL / 16, 256, 0, stream>>>(xre, xim, Ctre, Ctim, Pre, Pim, care, caim, y);
}